// GraphPositionalEncoding_40879498729274
// MI455X (gfx1250) — compile-verified
//
#include <hip/hip_runtime.h>

#define D_MODEL 1024
#define T_SEQ   512
#define BATCH   64
#define SPLIT   8           // t-slices per batch row -> 512 blocks total
#define TPB     512         // 16 waves (wave32) per block

// ---- CDNA5 async global->LDS path (guarded so compile never breaks) --------
#if defined(__has_builtin)
#  if __has_builtin(__builtin_amdgcn_global_load_async_to_lds_b128)
#    define GPE_HAVE_ASYNC 1
#  endif
#  if __has_builtin(__builtin_amdgcn_s_wait_asynccnt)
#    define GPE_HAVE_WAITASYNC 1
#  endif
#endif

typedef int   v4i __attribute__((vector_size(16)));
typedef float v4f __attribute__((vector_size(16)));
typedef v4i __attribute__((address_space(1))) *gv4i_ptr;  // global
typedef v4i __attribute__((address_space(3))) *lv4i_ptr;  // LDS

__device__ __forceinline__ void async_copy16(const float* g, float* l) {
#if defined(GPE_HAVE_ASYNC)
  // per-lane 16B async DMA: global (AS1) -> LDS (AS3), tracked by ASYNCcnt
  __builtin_amdgcn_global_load_async_to_lds_b128(
      (gv4i_ptr)g, (lv4i_ptr)l, /*imm offset*/0, /*cpol*/0);
#else
  *(v4f*)l = *(const v4f*)g;
#endif
}

__device__ __forceinline__ void wait_async_all() {
#if defined(GPE_HAVE_ASYNC)
#  if defined(GPE_HAVE_WAITASYNC)
  __builtin_amdgcn_s_wait_asynccnt(0);
#  else
  asm volatile("s_wait_asynccnt 0" ::: "memory");
#  endif
#endif
}

// ---------------------------------------------------------------------------
__global__ __launch_bounds__(TPB)
void GraphPositionalEncoding_40879498729274_kernel(
    const int*   __restrict__ input_ids,   // [B,T]
    const int*   __restrict__ rel_ids,     // [B,T]
    const float* __restrict__ seq_table,   // [512,1024]
    const float* __restrict__ chain_table, // [1000,1024]
    const float* __restrict__ depth_table, // [20,1024]
    const float* __restrict__ role_table,  // [4,1024]
    float*       __restrict__ out)         // [B,T,1024]
{
  __shared__ int s_scan[T_SEQ];
  __shared__ int s_base[T_SEQ];
  __shared__ int s_cp[T_SEQ];
  __shared__ int s_role[T_SEQ];
  __shared__ __align__(16) float s_raw[5 * D_MODEL];   // role[4096] ++ depth0[1024]
  __shared__ __align__(16) float s_fused[4 * D_MODEL]; // 0.2*role + 0.3*depth0

  const int tid    = threadIdx.x;
  const int b      = blockIdx.x / SPLIT;
  const int tslice = blockIdx.x % SPLIT;
  const int g0     = b * T_SEQ;

  // ---- Phase 0: kick off async staging of role_table + depth_table row 0 --
  // 1280 x 16B chunks: [0,1024) = role_table (16KB), [1024,1280) = depth row0
  for (int i = tid; i < (5 * D_MODEL) / 4; i += TPB) {
    const float* g = (i < D_MODEL) ? (role_table + 4 * i)
                                   : (depth_table + 4 * (i - D_MODEL));
    async_copy16(g, &s_raw[4 * i]);
  }

  // ---- Phase 1: per-row scan for chain positions (runs while DMA in flight)
  // c   = exclusive prefix sum of (rel > 0)
  // rst = (rel==0 && prev_rel>0);  baseline = cummax(rst ? c : 0)
  // cp  = c - baseline
  const int rel = rel_ids[g0 + tid];
  const int iid = input_ids[g0 + tid];
  const int inc = (rel > 0) ? 1 : 0;

  s_scan[tid] = inc;
  __syncthreads();
  #pragma unroll
  for (int o = 1; o < T_SEQ; o <<= 1) {
    int v = (tid >= o) ? s_scan[tid - o] : 0;
    __syncthreads();
    s_scan[tid] += v;
    __syncthreads();
  }
  const int c = s_scan[tid] - inc;               // exclusive count

  const int prev_rel = (tid > 0) ? rel_ids[g0 + tid - 1] : 0;
  s_base[tid] = (rel == 0 && prev_rel > 0) ? c : 0;
  __syncthreads();
  #pragma unroll
  for (int o = 1; o < T_SEQ; o <<= 1) {
    int v = (tid >= o) ? s_base[tid - o] : 0;    // inclusive cummax
    __syncthreads();
    if (v > s_base[tid]) s_base[tid] = v;
    __syncthreads();
  }
  s_cp[tid]   = c - s_base[tid];
  s_role[tid] = (iid <= 4) ? 3 : ((rel == 0) ? 2 : 0);

  // ---- Phase 2: wait for DMA, fuse 0.2*role + 0.3*depth0 in LDS -----------
  wait_async_all();          // each wave drains its own ASYNCcnt
  __syncthreads();           // ...then all waves' copies are visible
  for (int i = tid; i < 4 * D_MODEL; i += TPB) {
    s_fused[i] = 0.2f * s_raw[i] + 0.3f * s_raw[4 * D_MODEL + (i & (D_MODEL - 1))];
  }
  __syncthreads();

  // ---- Phase 3: stream the output slice (bandwidth-bound, 16B vectors) ---
  // Output is write-once: use non-temporal stores so the 128MB stream does
  // not evict the L2-resident seq/chain tables shared by all blocks.
  const int nT   = T_SEQ / SPLIT;       // 64 rows of t per block
  const int t0   = tslice * nT;
  const int d4   = tid & 255;           // 256 x 16B per d-row
  const int tofs = tid >> 8;            // 512 threads cover 2 t-rows per step

  const v4f* seq4   = (const v4f*)seq_table;
  const v4f* chain4 = (const v4f*)chain_table;
  const v4f* fused4 = (const v4f*)s_fused;
  v4f*       out4   = (v4f*)out;

  for (int ti = tofs; ti < nT; ti += 2) {
    const int t  = t0 + ti;
    const int cp = s_cp[t];
    const int ro = s_role[t];
    v4f sq = seq4[t * (D_MODEL / 4) + d4];
    v4f ch = chain4[cp * (D_MODEL / 4) + d4];
    v4f fu = fused4[ro * (D_MODEL / 4) + d4];
    v4f o  = sq + 0.5f * ch + fu;
    __builtin_nontemporal_store(o, &out4[((size_t)b * T_SEQ + t) * (D_MODEL / 4) + d4]);
  }
}

// ---------------------------------------------------------------------------
extern "C" void kernel_launch(void* const* d_in, const int* in_sizes, int n_in,
                              void* d_out, int out_size, void* d_ws, size_t ws_size,
                              hipStream_t stream) {
  (void)in_sizes; (void)n_in; (void)out_size; (void)d_ws; (void)ws_size;
  const int*   input_ids   = (const int*)d_in[0];
  const int*   rel_ids     = (const int*)d_in[1];
  const float* seq_table   = (const float*)d_in[2];
  const float* chain_table = (const float*)d_in[3];
  const float* depth_table = (const float*)d_in[4];
  const float* role_table  = (const float*)d_in[5];
  float* out = (float*)d_out;

  dim3 grid(BATCH * SPLIT);   // 512 blocks
  dim3 block(TPB);            // 512 threads = 16 wave32
  GraphPositionalEncoding_40879498729274_kernel<<<grid, block, 0, stream>>>(
      input_ids, rel_ids, seq_table, chain_table, depth_table, role_table, out);
}